// Model_37306085933426
// MI455X (gfx1250) — compile-verified
//
#include <hip/hip_runtime.h>

typedef float v2f __attribute__((ext_vector_type(2)));
typedef float v8f __attribute__((ext_vector_type(8)));

#define OUTN   224
#define IMH    768
#define IMW    1024
#define HW     (768 * 1024)
#define KDIM   768          // 3*16*16
#define NPOS   196          // 14*14 patches
#define FEAT   18816        // 96*196
#define NWAVE  13           // 13 M-tiles of 16 (196 -> 208 padded)

__launch_bounds__(416, 1)
__global__ void crop_embed_kernel(const float* __restrict__ image_pair,
                                  const float* __restrict__ ref_pts,
                                  const float* __restrict__ ip0,
                                  const float* __restrict__ ip1,
                                  const float* __restrict__ conv_w,
                                  const float* __restrict__ conv_b,
                                  float* __restrict__ out)
{
    __shared__ int   s_ix[OUTN];
    __shared__ int   s_iy[OUTN];
    __shared__ float s_part[16];
    __shared__ float s_scale;

    const int g   = blockIdx.x;   // 0..191 crop id, matches output order z1,z2,y1,y2
    const int tid = threadIdx.x;

    // ---- resolve image + point for this crop ----
    int b, pair;
    const float* pt;
    if (g < 64)        { int h = g;       b = h >> 5; pair = 0; pt = ref_pts + (((size_t)b*2 + 0)*64 + (h & 31))*2; }
    else if (g < 128)  { int h = g - 64;  b = h >> 5; pair = 1; pt = ref_pts + (((size_t)b*2 + 1)*64 + (h & 31))*2; }
    else if (g < 160)  { int h = g - 128; b = h >> 4; pair = 0; pt = ip0 + ((size_t)b*32 + (h & 15))*2; }
    else               { int h = g - 160; b = h >> 4; pair = 1; pt = ip1 + ((size_t)b*32 + (h & 15))*2; }
    const float* img = image_pair + (size_t)(b*6 + pair*3) * HW;

    // ---- crop index tables (nearest gather), semantics match JAX int32 cast+clip ----
    if (tid < OUTN) {
        float px = pt[0] * 1024.0f;
        float py = pt[1] * 768.0f;
        int x_min = (int)(px - 32.0f); x_min = x_min < 0 ? 0 : (x_min > IMW ? IMW : x_min);
        int x_max = (int)(px + 32.0f); x_max = x_max < 0 ? 0 : (x_max > IMW ? IMW : x_max);
        int y_min = (int)(py - 32.0f); y_min = y_min < 0 ? 0 : (y_min > IMH ? IMH : y_min);
        int y_max = (int)(py + 64.0f); y_max = y_max < 0 ? 0 : (y_max > IMH ? IMH : y_max);
        int cw = x_max - x_min;
        int ch = y_max - y_min;
        s_ix[tid] = x_min + (tid * cw) / OUTN;
        s_iy[tid] = y_min + (tid * ch) / OUTN;
    }
    __syncthreads();

    // ---- per-lane geometry ----
    // Rows m >= 196 are padding. Their D rows are discarded post-GEMM, so we
    // simply clamp their addresses in-bounds (duplicate patch 195) instead of
    // masking loads -> zero divergence in the hot loop.
    const int lane   = tid & 31;
    const int wave   = tid >> 5;        // 0..12 == M-tile index
    const int khalf  = lane >> 4;       // 0: K pair {0,1}; 1: K pair {2,3}
    const int lm     = lane & 15;
    const int m      = wave * 16 + lm;  // A-matrix row (patch id) for this lane
    const int mc     = m < (NPOS - 1) ? m : (NPOS - 1);   // clamped for addressing
    const int ty = mc / 14;
    const int tx = mc - ty * 14;
    const int iyBase = ty * 16;
    const int ixBase = tx * 16;

    // Preload the 8 column indices this lane ever needs (k0 in {0,4,8,12}, pair at khalf*2)
    int ixv[8];
#pragma unroll
    for (int q = 0; q < 4; ++q) {
        int base = ixBase + q * 4 + khalf * 2;
        ixv[2*q]     = s_ix[base];
        ixv[2*q + 1] = s_ix[base + 1];
    }

    v8f acc[6];
    v8f zero = {};
#pragma unroll
    for (int t = 0; t < 6; ++t) acc[t] = zero;

    // ---- GEMM: X(196x768) * W^T(768x96), K = c*256 + ky*16 + kx, via f32 WMMA 16x16x4 ----
#pragma unroll 1
    for (int c = 0; c < 3; ++c) {
        const float* chan = img + (size_t)c * HW;
#pragma unroll 1
        for (int ky = 0; ky < 16; ++ky) {
            const int iyv = s_iy[iyBase + ky];
            const float* row = chan + (size_t)iyv * IMW;
            const int kbase = c * 256 + ky * 16;
#pragma unroll
            for (int q = 0; q < 4; ++q) {
                v2f a;
                a.x = row[ixv[2*q]];
                a.y = row[ixv[2*q + 1]];
                const int k0 = kbase + q * 4 + khalf * 2;   // even -> 8B aligned B loads
                const float* wrow = conv_w + (size_t)lm * KDIM + k0;
#pragma unroll
                for (int t = 0; t < 6; ++t) {
                    v2f bf = *(const v2f*)(wrow + t * 16 * KDIM);
                    acc[t] = __builtin_amdgcn_wmma_f32_16x16x4_f32(
                        false, a, false, bf, (short)0, acc[t], false, false);
                }
            }
        }
    }

    // ---- bias + ReLU + local sum of squares (D rows: m = wave*16 + khalf*8 + r) ----
    const int mrow0 = wave * 16 + khalf * 8;
    float sumsq = 0.0f;
#pragma unroll
    for (int t = 0; t < 6; ++t) {
        const float bias = conv_b[t * 16 + lm];
#pragma unroll
        for (int r = 0; r < 8; ++r) {
            float v = acc[t][r] + bias;
            v = v > 0.0f ? v : 0.0f;
            if (mrow0 + r >= NPOS) v = 0.0f;    // discard padded rows
            acc[t][r] = v;
            sumsq += v * v;
        }
    }

    // deterministic reduction: butterfly within wave32, then serial over 13 waves
#pragma unroll
    for (int off = 16; off > 0; off >>= 1)
        sumsq += __shfl_xor(sumsq, off, 32);
    if (lane == 0) s_part[wave] = sumsq;
    __syncthreads();
    if (tid == 0) {
        float tot = 0.0f;
        for (int wv = 0; wv < NWAVE; ++wv) tot += s_part[wv];
        float nrm = sqrtf(tot);
        s_scale = 1.0f / fmaxf(nrm, 1e-12f);
    }
    __syncthreads();
    const float scale = s_scale;

    // ---- scaled store: out[g*18816 + n*196 + m] ----
    float* og = out + (size_t)g * FEAT;
#pragma unroll
    for (int t = 0; t < 6; ++t) {
        float* oc = og + (t * 16 + lm) * NPOS;
#pragma unroll
        for (int r = 0; r < 8; ++r) {
            int mr = mrow0 + r;
            if (mr < NPOS) oc[mr] = acc[t][r] * scale;
        }
    }
}

extern "C" void kernel_launch(void* const* d_in, const int* in_sizes, int n_in,
                              void* d_out, int out_size, void* d_ws, size_t ws_size,
                              hipStream_t stream) {
    const float* image_pair = (const float*)d_in[0];
    const float* ref_pts    = (const float*)d_in[1];
    const float* ip0        = (const float*)d_in[2];
    const float* ip1        = (const float*)d_in[3];
    const float* conv_w     = (const float*)d_in[4];
    const float* conv_b     = (const float*)d_in[5];
    // d_in[6..8]: ref_num=32, independ_num0=16, independ_num1=16 (fixed by harness)
    float* out = (float*)d_out;
    hipLaunchKernelGGL(crop_embed_kernel, dim3(192), dim3(416), 0, stream,
                       image_pair, ref_pts, ip0, ip1, conv_w, conv_b, out);
}